// RoIPooling_26130581028992
// MI455X (gfx1250) — compile-verified
//
#include <hip/hip_runtime.h>
#include <hip/hip_bf16.h>
#include <math.h>

// RoI max pooling for MI455X (gfx1250, wave32).
// features: (B=2, C=256, H=38, W=38) fp32 ; rois: (64, 5) int32
// out: (64, 256, 7, 7) fp32
//
// Problem is L2-resident (features = ~3 MB << 192 MB L2) => latency-bound.
// Design: one thread per output element; 49 blocks x 256 threads per RoI so
// the RoI descriptor is block-uniform (scalar loads); ow fastest for W-contiguous
// access; global_prefetch_b8 to warm the window's cacheline.

#define ROI_C       256
#define ROI_H       38
#define ROI_W       38
#define ROI_OUT_H   7
#define ROI_OUT_W   7
#define ROI_NBINS   (ROI_OUT_H * ROI_OUT_W)       // 49
#define ROI_SCALE   0.0625f
#define N_ROIS      64
#define BLOCKS_PER_ROI 49                          // 256*49 == 256 ch * 49 bins

__global__ __launch_bounds__(256)
void roi_max_pool_kernel(const float* __restrict__ feat,
                         const int*   __restrict__ rois,
                         float*       __restrict__ out)
{
    // Block-uniform RoI index: enables scalar (s_load) RoI descriptor reads.
    const int r     = blockIdx.x / BLOCKS_PER_ROI;
    const int local = (blockIdx.x % BLOCKS_PER_ROI) * 256 + threadIdx.x;

    const int c   = local / ROI_NBINS;   // channel 0..255
    const int bin = local % ROI_NBINS;   // 0..48
    const int oh  = bin / ROI_OUT_W;
    const int ow  = bin % ROI_OUT_W;

    // RoI descriptor (uniform across the block -> SGPRs).
    const int   bidx    = rois[r * 5 + 0];
    const float start_w = (float)rois[r * 5 + 1] * ROI_SCALE;
    const float start_h = (float)rois[r * 5 + 2] * ROI_SCALE;
    const float end_w   = (float)rois[r * 5 + 3] * ROI_SCALE;
    const float end_h   = (float)rois[r * 5 + 4] * ROI_SCALE;

    const float bin_h = fmaxf(end_h - start_h, 1.0f) / (float)ROI_OUT_H;
    const float bin_w = fmaxf(end_w - start_w, 1.0f) / (float)ROI_OUT_W;

    // Same fp32 math/order as the reference, clip to [0, H] / [0, W].
    int hs = (int)floorf((float)oh * bin_h + start_h);
    int he = (int)ceilf(((float)oh + 1.0f) * bin_h + start_h);
    int ws = (int)floorf((float)ow * bin_w + start_w);
    int we = (int)ceilf(((float)ow + 1.0f) * bin_w + start_w);
    hs = min(max(hs, 0), ROI_H);
    he = min(max(he, 0), ROI_H);
    ws = min(max(ws, 0), ROI_W);
    we = min(max(we, 0), ROI_W);

    const float* __restrict__ plane =
        feat + ((size_t)bidx * ROI_C + (size_t)c) * (ROI_H * ROI_W);

    // gfx1250 prefetch (global_prefetch_b8): warm the first window line.
    __builtin_prefetch(plane + hs * ROI_W + ws, 0, 1);

    float v = -INFINITY;
    for (int h = hs; h < he; ++h) {
        const float* __restrict__ row = plane + h * ROI_W;
        #pragma unroll 4
        for (int w = ws; w < we; ++w) {
            v = fmaxf(v, row[w]);
        }
    }
    // Empty window in either dimension -> 0 (matches reference `valid` mask).
    if ((he <= hs) || (we <= ws)) v = 0.0f;

    out[((size_t)r * ROI_C + (size_t)c) * ROI_NBINS + (size_t)bin] = v;
}

extern "C" void kernel_launch(void* const* d_in, const int* in_sizes, int n_in,
                              void* d_out, int out_size, void* d_ws, size_t ws_size,
                              hipStream_t stream)
{
    (void)in_sizes; (void)n_in; (void)d_ws; (void)ws_size; (void)out_size;

    const float* feat = (const float*)d_in[0];   // (2,256,38,38) fp32
    const int*   rois = (const int*)d_in[1];     // (64,5) int32
    float*       out  = (float*)d_out;           // (64,256,7,7) fp32

    dim3 grid(N_ROIS * BLOCKS_PER_ROI);          // 3136 blocks
    dim3 block(256);                             // 8 wave32s per block
    roi_max_pool_kernel<<<grid, block, 0, stream>>>(feat, rois, out);
}